// HaarWaveletTopK_6339371729046
// MI455X (gfx1250) — compile-verified
//
#include <hip/hip_runtime.h>
#include <stdint.h>

#define BB 8
#define TT_TOT 8192
#define FF 512
#define T2 4096
#define NCHUNK 32          // t-chunks per batch
#define CT 128             // t-pairs per chunk
#define TILE_T 8           // t-pairs per LDS tile
#define NSTEP (CT / TILE_T) // 16 steps
#define TILE_FLOATS (TILE_T * 2 * FF)   // 8192 floats = 32KB
#define TILE_BYTES (TILE_FLOATS * 4)

typedef unsigned long long u64;
typedef uint32_t u32;

// ---- CDNA5 async tensor-path: global -> LDS DMA (ASYNCcnt tracked) ----
__device__ __forceinline__ void async_b128(u32 lds_off, const void* sbase, u32 voff) {
    // GVS form: LDS address in VDST, global address SADDR + VADDR; tracked by ASYNCcnt
    asm volatile("global_load_async_to_lds_b128 %0, %1, %2"
                 :: "v"(lds_off), "v"(voff), "s"(sbase) : "memory");
}
__device__ __forceinline__ void wait_async_le4() {
    asm volatile("s_wait_asynccnt 4" ::: "memory");
}
__device__ __forceinline__ void wait_async_0() {
    asm volatile("s_wait_asynccnt 0" ::: "memory");
}

// Kernel A: stream Haar transform + zero-fill detail + per-chunk top-8 partials.
// grid = (NCHUNK, BB), block = 512 (thread tid owns column f = tid).
__global__ __launch_bounds__(512) void haar_stream_topk(
    const float* __restrict__ x, float* __restrict__ mainp,
    float* __restrict__ detail, u64* __restrict__ partial) {
    __shared__ float sm[2][TILE_FLOATS];   // 2 x 32KB double buffer

    const int tid   = threadIdx.x;
    const int chunk = blockIdx.x;
    const int b     = blockIdx.y;

    // LDS byte offset of the shared buffer (low 32 bits of generic address)
    const u32 smbase = (u32)(uintptr_t)(&sm[0][0]);
    // this block covers rows [2*chunk*CT, 2*(chunk+1)*CT) of batch b: contiguous 512KB
    const char* gbase = (const char*)(x + ((size_t)b * TT_TOT + (size_t)chunk * CT * 2) * FF);

    u64 c[8];                      // sorted descending top-8 keys
    #pragma unroll
    for (int i = 0; i < 8; ++i) c[i] = 0ull;

    // prefetch tile 0 into buffer 0 (4 x b128 per lane = 32KB per block)
    {
        const u32 lo = smbase + (u32)tid * 16u;
        const u32 go = (u32)tid * 16u;
        #pragma unroll
        for (int i = 0; i < 4; ++i) async_b128(lo + i * 8192u, gbase, go + i * 8192u);
    }

    for (int s = 0; s < NSTEP; ++s) {
        const int cur = s & 1;
        if (s + 1 < NSTEP) {
            const u32 lo = smbase + (u32)(cur ^ 1) * TILE_BYTES + (u32)tid * 16u;
            const u32 go = (u32)(s + 1) * TILE_BYTES + (u32)tid * 16u;
            #pragma unroll
            for (int i = 0; i < 4; ++i) async_b128(lo + i * 8192u, gbase, go + i * 8192u);
            wait_async_le4();      // oldest 4 (current tile) complete, next 4 in flight
        } else {
            wait_async_0();
        }
        __syncthreads();           // all waves' async loads for 'cur' are visible

        #pragma unroll
        for (int j = 0; j < TILE_T; ++j) {
            const float e = sm[cur][(2 * j) * FF + tid];
            const float o = sm[cur][(2 * j + 1) * FF + tid];
            const float low = (e + o) * 0.5f;          // x_low * inv_sqrt2
            const u32 t = (u32)chunk * CT + (u32)s * TILE_T + (u32)j;  // t-pair idx
            const size_t row = ((size_t)b * TT_TOT + 2 * (size_t)t) * FF + tid;
            __builtin_nontemporal_store(low,  &mainp[row]);
            __builtin_nontemporal_store(low,  &mainp[row + FF]);
            __builtin_nontemporal_store(0.0f, &detail[row]);
            __builtin_nontemporal_store(0.0f, &detail[row + FF]);
            // monotonic key: magnitude bits in high word, t in low word
            const u32 hb = __float_as_uint(fabsf(e - o));
            const u64 key = ((u64)hb << 32) | (u64)t;
            if (key > c[7]) {                          // insert keeping descending order
                c[7] = key;
                #pragma unroll
                for (int i = 7; i > 0; --i) {
                    if (c[i] > c[i - 1]) { u64 tmp = c[i - 1]; c[i - 1] = c[i]; c[i] = tmp; }
                }
            }
        }
        __syncthreads();           // tile fully consumed before it is overwritten
    }

    // contiguous 64B store per thread: partial[((b*32+chunk)*512+f)*8 + i]
    const size_t pbase = (((size_t)b * NCHUNK + chunk) * FF + tid) * 8;
    #pragma unroll
    for (int i = 0; i < 8; ++i) partial[pbase + i] = c[i];
}

// Kernel B: one wave32 per (b,f) column; merge 32 sorted 8-lists -> top-8,
// then scatter the 8 nonzero detail pairs. grid = 512, block = 256 (8 waves).
__global__ __launch_bounds__(256) void haar_merge_scatter(
    const float* __restrict__ x, float* __restrict__ detail,
    const u64* __restrict__ partial) {
    const int lane = threadIdx.x & 31;     // lane == chunk index
    const int wid  = threadIdx.x >> 5;
    const int col  = blockIdx.x * 8 + wid; // 0..4095
    const int b = col >> 9;
    const int f = col & (FF - 1);

    u64 c[8];
    const size_t pbase = (((size_t)b * NCHUNK + lane) * FF + f) * 8;
    #pragma unroll
    for (int i = 0; i < 8; ++i) c[i] = partial[pbase + i];

    int rem = 8;
    u64 mykey = 0ull;
    #pragma unroll
    for (int r = 0; r < 8; ++r) {
        const u64 prop = (rem > 0) ? c[0] : 0ull;
        u64 m = prop;
        #pragma unroll
        for (int off = 16; off > 0; off >>= 1) {   // wave32 max-reduce
            const u64 ov = __shfl_xor(m, off, 32);
            if (ov > m) m = ov;
        }
        if (prop == m && rem > 0 && m != 0ull) {   // unique winner pops its head
            #pragma unroll
            for (int i = 0; i < 7; ++i) c[i] = c[i + 1];
            c[7] = 0ull;
            --rem;
        }
        if (lane == r) mykey = m;                  // lane r owns round-r winner
    }

    if (lane < 8 && mykey != 0ull) {
        const u32 t = (u32)(mykey & 0xffffffffull);
        const size_t row = ((size_t)b * TT_TOT + 2 * (size_t)t) * FF + f;
        const float e = x[row];
        const float o = x[row + FF];
        const float hh = (e - o) * 0.5f;           // x_high * inv_sqrt2
        detail[row]      = hh;
        detail[row + FF] = -hh;
    }
}

extern "C" void kernel_launch(void* const* d_in, const int* in_sizes, int n_in,
                              void* d_out, int out_size, void* d_ws, size_t ws_size,
                              hipStream_t stream) {
    const float* x = (const float*)d_in[0];
    float* mainp  = (float*)d_out;
    float* detail = mainp + (size_t)BB * TT_TOT * FF;   // second tuple element
    u64* partial  = (u64*)d_ws;                         // needs 8 MB

    dim3 gA(NCHUNK, BB);
    haar_stream_topk<<<gA, 512, 0, stream>>>(x, mainp, detail, partial);

    const int ncols = BB * FF;                          // 4096 columns
    haar_merge_scatter<<<ncols / 8, 256, 0, stream>>>(x, detail, partial);
}